// DiscriminativeLoss_3882650436520
// MI455X (gfx1250) — compile-verified
//
#include <hip/hip_runtime.h>
#include <hip/hip_bf16.h>

// ---------------------------------------------------------------------------
// DiscriminativeLoss forward for MI455X (gfx1250, wave32, WMMA).
//   N=1024 features of D=512 f32, C=64 multilabels, n_sim = 523 mined pairs.
// Pipeline:
//   1) row norms sq[i]
//   2) WMMA f32 16x16x4 Gram -> d2[i,j] = sq_i + sq_j - 2*F.F^T   (N x N)
//      (B tile staged in LDS via GLOBAL_LOAD_ASYNC_TO_LDS_B128)
//   3) agree[i,j] = 1 - 0.5*L1(ml_i, ml_j)   (tiles staged via async-to-LDS)
//   4) two-level histogram -> agreement threshold (523rd largest, from top)
//   5) two-level histogram -> distance cutoff (523rd smallest, from bottom)
//   6) masked reduction over upper triangle -> 4 accumulators
//   7) scalar loss
// ---------------------------------------------------------------------------

#define NN   1024
#define DD   512
#define CC   64
#define NBIN 4096
#define NSIM 523

typedef __attribute__((ext_vector_type(2))) float v2f;
typedef __attribute__((ext_vector_type(8))) float v8f;

// ---- CDNA5 async memory->LDS helpers --------------------------------------
// GVS mode: mem_addr = SGPR64 base + VGPR32 offset ; LDS dest addr in VGPR.
// Generic pointers to LDS carry the workgroup-relative offset in their low
// 32 bits (ISA 10.2), so truncation is valid.
__device__ __forceinline__ unsigned lds_off32(const void* p) {
  return (unsigned)(unsigned long long)p;
}

__device__ __forceinline__ void async_load_b128(unsigned lds_addr,
                                                const void* gbase,
                                                unsigned goff) {
  asm volatile("global_load_async_to_lds_b128 %0, %1, %2"
               :: "v"(lds_addr), "v"(goff), "s"(gbase)
               : "memory");
}

__device__ __forceinline__ void wait_asynccnt0() {
#if __has_builtin(__builtin_amdgcn_s_wait_asynccnt)
  __builtin_amdgcn_s_wait_asynccnt(0);
#else
  asm volatile("s_wait_asynccnt 0x0" ::: "memory");
#endif
}

// ---------------- workspace layout (bytes) ----------------
static const size_t OFF_D2    = 0;
static const size_t OFF_AG    = (size_t)NN * NN * 4;
static const size_t OFF_SQ    = OFF_AG + (size_t)NN * NN * 4;
static const size_t OFF_HIST  = OFF_SQ + 4096;
static const size_t OFF_STATE = OFF_HIST + NBIN * 4;

// ---------------------------------------------------------------------------
__global__ __launch_bounds__(256) void zero_u32_kernel(unsigned* p, int n) {
  int i = blockIdx.x * blockDim.x + threadIdx.x;
  if (i < n) p[i] = 0u;
}

// ---------------------------------------------------------------------------
// sq[i] = sum_k F[i,k]^2 ; one wave32 per row, 8 rows per block
__global__ __launch_bounds__(256) void rownorm_kernel(const float* __restrict__ F,
                                                      float* __restrict__ sq) {
  int row  = blockIdx.x * 8 + (threadIdx.x >> 5);
  int lane = threadIdx.x & 31;
  const float4* r4 = (const float4*)(F + (size_t)row * DD);
  float s = 0.f;
  for (int c = lane; c < DD / 4; c += 32) {
    float4 v = r4[c];
    s += v.x * v.x + v.y * v.y + v.z * v.z + v.w * v.w;
  }
  for (int off = 16; off > 0; off >>= 1) s += __shfl_down(s, off, 32);
  if (lane == 0) sq[row] = s;
}

// ---------------------------------------------------------------------------
// Gram + distance kernel: block = 8 waves; each wave owns one 16x16 tile of G.
// Block covers rows [i0,i0+128), cols [j0,j0+16). The 16 B-rows (full K=512)
// are DMA'd into LDS once (async-to-LDS, no VGPR round trip) and reused by
// all 8 waves. WMMA f32 16x16x4 accumulates over K in steps of 4.
__global__ __launch_bounds__(256) void gram_d2_kernel(const float* __restrict__ F,
                                                      const float* __restrict__ sq,
                                                      float* __restrict__ d2) {
  __shared__ float Bs[16 * DD];            // 32 KB
  const int j0   = blockIdx.x * 16;
  const int i0   = blockIdx.y * 128;
  const int wave = threadIdx.x >> 5;
  const int lane = threadIdx.x & 31;

  // async DMA of the 16 col-rows of F into LDS: 2048 x b128, 8 per thread
  {
    const void* gbase = (const void*)(F + (size_t)j0 * DD);
    const unsigned lbase = lds_off32(Bs);
    for (int t = threadIdx.x; t < 16 * DD / 4; t += 256)
      async_load_b128(lbase + t * 16u, gbase, t * 16u);
    wait_asynccnt0();
  }
  __syncthreads();

  // A 16x4 f32 fragment layout: lanes 0-15 -> M=lane, K=k0+{0,1};
  //                             lanes 16-31 -> M=lane-16, K=k0+{2,3}.
  // B 4x16 fragment is the symmetric layout on the column tile.
  const int lane_m    = lane & 15;
  const int lane_koff = (lane >> 4) * 2;
  const int row_base  = i0 + wave * 16;

  const float* Arow = F + (size_t)(row_base + lane_m) * DD + lane_koff;
  const float* Brow = Bs + lane_m * DD + lane_koff;
  __builtin_prefetch(Arow, 0, 1);          // global_prefetch_b8 into L2/L0

  v8f acc = {};
  for (int k = 0; k < DD; k += 4) {
    v2f a = *(const v2f*)(Arow + k);       // global_load_b64
    v2f b = *(const v2f*)(Brow + k);       // ds_load_b64
    acc = __builtin_amdgcn_wmma_f32_16x16x4_f32(
        /*neg_a=*/false, a, /*neg_b=*/false, b,
        /*c_mod=*/(short)0, acc, /*reuse_a=*/false, /*reuse_b=*/false);
  }

  // C/D layout: VGPR r -> lanes 0-15: M=r ; lanes 16-31: M=r+8 ; N=lane&15
  const int col  = j0 + lane_m;
  const int row0 = row_base + ((lane >> 4) << 3);
  const float sqc = sq[col];
  for (int r = 0; r < 8; ++r) {
    int row = row0 + r;
    d2[(size_t)row * NN + col] = sq[row] + sqc - 2.0f * acc[r];
  }
}

// ---------------------------------------------------------------------------
// agree[i,j] = 1 - 0.5 * sum_c |ml[i,c]-ml[j,c]| ; 32x32 tiles via async DMA
__global__ __launch_bounds__(256) void agree_kernel(const float* __restrict__ ML,
                                                    float* __restrict__ agree) {
  __shared__ float As[32 * CC];
  __shared__ float Bs[32 * CC];
  const int ti = blockIdx.y * 32;
  const int tj = blockIdx.x * 32;
  {
    const void* gA = (const void*)(ML + (size_t)ti * CC);
    const void* gB = (const void*)(ML + (size_t)tj * CC);
    const unsigned lA = lds_off32(As);
    const unsigned lB = lds_off32(Bs);
    for (int t = threadIdx.x; t < 32 * CC / 4; t += 256) {
      async_load_b128(lA + t * 16u, gA, t * 16u);
      async_load_b128(lB + t * 16u, gB, t * 16u);
    }
    wait_asynccnt0();
  }
  __syncthreads();
  for (int p = threadIdx.x; p < 32 * 32; p += 256) {
    int li = p >> 5, lj = p & 31;
    const float4* a4 = (const float4*)(As + li * CC);
    const float4* b4 = (const float4*)(Bs + lj * CC);
    float s = 0.f;
    for (int c = 0; c < CC / 4; ++c) {
      float4 a = a4[c], b = b4[c];
      s += fabsf(a.x - b.x) + fabsf(a.y - b.y) + fabsf(a.z - b.z) + fabsf(a.w - b.w);
    }
    agree[(size_t)(ti + li) * NN + (tj + lj)] = 1.0f - 0.5f * s;
  }
}

// ---------------------------------------------------------------------------
// LDS-privatized histogram over upper-triangle values of M.
// fine==0: coarse bins over [vmin,vmax). fine==1: sub-bins inside coarse bin.
__global__ __launch_bounds__(256) void hist_kernel(const float* __restrict__ M,
                                                   unsigned* __restrict__ hist,
                                                   const unsigned* __restrict__ stateu,
                                                   int fine, float vmin, float vmax) {
  __shared__ unsigned h[NBIN];
  for (int t = threadIdx.x; t < NBIN; t += 256) h[t] = 0u;
  __syncthreads();

  const float inv   = (float)NBIN / (vmax - vmin);
  const float width = (vmax - vmin) / (float)NBIN;
  unsigned cb = 0u; float flo = 0.f, finv = 0.f;
  if (fine) { cb = stateu[0]; flo = vmin + (float)cb * width; finv = (float)NBIN / width; }

  const size_t total = (size_t)NN * NN;
  for (size_t idx = (size_t)blockIdx.x * blockDim.x + threadIdx.x; idx < total;
       idx += (size_t)gridDim.x * blockDim.x) {
    int i = (int)(idx >> 10), j = (int)(idx & (NN - 1));
    if (j <= i) continue;
    float v = M[idx];
    int b = (int)((v - vmin) * inv);
    b = b < 0 ? 0 : (b > NBIN - 1 ? NBIN - 1 : b);
    if (!fine) {
      atomicAdd(&h[b], 1u);
    } else if ((unsigned)b == cb) {
      int s = (int)((v - flo) * finv);
      s = s < 0 ? 0 : (s > NBIN - 1 ? NBIN - 1 : s);
      atomicAdd(&h[s], 1u);
    }
  }
  __syncthreads();
  for (int t = threadIdx.x; t < NBIN; t += 256)
    if (h[t]) atomicAdd(&hist[t], h[t]);
}

// ---------------------------------------------------------------------------
// Single-thread histogram scan. from_top=1: k-th largest ; 0: k-th smallest.
// Coarse pass stores {bin, cum_before} in stateu; fine pass writes threshold.
__global__ void scan_kernel(const unsigned* __restrict__ hist,
                            unsigned* __restrict__ stateu,
                            float* __restrict__ thr_out,
                            int from_top, int fine, int target,
                            float vmin, float vmax) {
  if (threadIdx.x != 0 || blockIdx.x != 0) return;
  const float width = (vmax - vmin) / (float)NBIN;
  unsigned cum = fine ? stateu[1] : 0u;
  if (!fine) {
    if (from_top) {
      for (int b = NBIN - 1; b >= 0; --b) {
        unsigned c = cum + hist[b];
        if (c >= (unsigned)target) { stateu[0] = (unsigned)b; stateu[1] = cum; return; }
        cum = c;
      }
      stateu[0] = 0u; stateu[1] = 0u;
    } else {
      for (int b = 0; b < NBIN; ++b) {
        unsigned c = cum + hist[b];
        if (c >= (unsigned)target) { stateu[0] = (unsigned)b; stateu[1] = cum; return; }
        cum = c;
      }
      stateu[0] = NBIN - 1; stateu[1] = 0u;
    }
  } else {
    const float lo = vmin + (float)stateu[0] * width;
    const float sw = width / (float)NBIN;
    if (from_top) {
      for (int s = NBIN - 1; s >= 0; --s) {
        cum += hist[s];
        if (cum >= (unsigned)target) { thr_out[0] = lo + (float)s * sw; return; }
      }
      thr_out[0] = lo;
    } else {
      for (int s = 0; s < NBIN; ++s) {
        cum += hist[s];
        if (cum >= (unsigned)target) { thr_out[0] = lo + (float)(s + 1) * sw; return; }
      }
      thr_out[0] = lo + width;
    }
  }
}

// ---------------------------------------------------------------------------
// Reduction over selected pairs: d2 <= cut ; positive if agree > thr.
// acc = [sum_e_pos, sum_e_neg, n_pos, n_neg]
__global__ __launch_bounds__(256) void accum_kernel(const float* __restrict__ d2,
                                                    const float* __restrict__ agree,
                                                    const float* __restrict__ thr_cut,
                                                    float* __restrict__ acc) {
  const float thr = thr_cut[0];
  const float cut = thr_cut[1];
  float sp = 0.f, sn = 0.f, np = 0.f, nn = 0.f;
  const size_t total = (size_t)NN * NN;
  for (size_t idx = (size_t)blockIdx.x * blockDim.x + threadIdx.x; idx < total;
       idx += (size_t)gridDim.x * blockDim.x) {
    int i = (int)(idx >> 10), j = (int)(idx & (NN - 1));
    if (j <= i) continue;
    float dv = d2[idx];
    if (dv <= cut) {
      float e = expf(-dv);
      if (agree[idx] > thr) { sp += e; np += 1.f; }
      else                  { sn += e; nn += 1.f; }
    }
  }
  __shared__ float red[4][256];
  red[0][threadIdx.x] = sp; red[1][threadIdx.x] = sn;
  red[2][threadIdx.x] = np; red[3][threadIdx.x] = nn;
  __syncthreads();
  for (int off = 128; off > 0; off >>= 1) {
    if (threadIdx.x < off)
      for (int q = 0; q < 4; ++q)
        red[q][threadIdx.x] += red[q][threadIdx.x + off];
    __syncthreads();
  }
  if (threadIdx.x == 0)
    for (int q = 0; q < 4; ++q) atomicAdd(&acc[q], red[q][0]);
}

// ---------------------------------------------------------------------------
__global__ void final_kernel(const float* __restrict__ acc, float* __restrict__ out) {
  if (threadIdx.x != 0 || blockIdx.x != 0) return;
  float sp = acc[0], sn = acc[1], np = acc[2], nn = acc[3];
  float pos_exp = (np > 0.f) ? sp / fmaxf(np, 1.f) : 1.0f;
  float num     = (np > 0.f) ? -logf(pos_exp) : 0.0f;
  float neg_exp = (nn > 0.f) ? sn / fmaxf(nn, 1.f) : 0.5f;
  out[0] = num + logf(pos_exp + neg_exp);
}

// ---------------------------------------------------------------------------
extern "C" void kernel_launch(void* const* d_in, const int* in_sizes, int n_in,
                              void* d_out, int out_size, void* d_ws, size_t ws_size,
                              hipStream_t stream) {
  (void)in_sizes; (void)n_in; (void)out_size; (void)ws_size;
  const float* F  = (const float*)d_in[0];   // [1024,512] f32
  const float* ML = (const float*)d_in[1];   // [1024,64] f32
  // d_in[2] = labels, unused by the loss
  float* out = (float*)d_out;

  char* ws = (char*)d_ws;
  float*    d2     = (float*)(ws + OFF_D2);
  float*    ag     = (float*)(ws + OFF_AG);
  float*    sq     = (float*)(ws + OFF_SQ);
  unsigned* hist   = (unsigned*)(ws + OFF_HIST);
  float*    statef = (float*)(ws + OFF_STATE);
  unsigned* stateu = (unsigned*)(ws + OFF_STATE + 64);

  // 1) row norms
  rownorm_kernel<<<NN / 8, 256, 0, stream>>>(F, sq);
  // 2) WMMA Gram -> squared distances
  gram_d2_kernel<<<dim3(NN / 16, NN / 128), 256, 0, stream>>>(F, sq, d2);
  // 3) pairwise agreement
  agree_kernel<<<dim3(NN / 32, NN / 32), 256, 0, stream>>>(ML, ag);

  // 4) agreement threshold: 523rd largest over [0,1]
  zero_u32_kernel<<<NBIN / 256, 256, 0, stream>>>(hist, NBIN);
  hist_kernel<<<512, 256, 0, stream>>>(ag, hist, stateu, 0, 0.f, 1.f);
  scan_kernel<<<1, 1, 0, stream>>>(hist, stateu, &statef[0], 1, 0, NSIM, 0.f, 1.f);
  zero_u32_kernel<<<NBIN / 256, 256, 0, stream>>>(hist, NBIN);
  hist_kernel<<<512, 256, 0, stream>>>(ag, hist, stateu, 1, 0.f, 1.f);
  scan_kernel<<<1, 1, 0, stream>>>(hist, stateu, &statef[0], 1, 1, NSIM, 0.f, 1.f);

  // 5) distance cutoff: 523rd smallest over [0,4096]
  zero_u32_kernel<<<NBIN / 256, 256, 0, stream>>>(hist, NBIN);
  hist_kernel<<<512, 256, 0, stream>>>(d2, hist, stateu, 0, 0.f, 4096.f);
  scan_kernel<<<1, 1, 0, stream>>>(hist, stateu, &statef[1], 0, 0, NSIM, 0.f, 4096.f);
  zero_u32_kernel<<<NBIN / 256, 256, 0, stream>>>(hist, NBIN);
  hist_kernel<<<512, 256, 0, stream>>>(d2, hist, stateu, 1, 0.f, 4096.f);
  scan_kernel<<<1, 1, 0, stream>>>(hist, stateu, &statef[1], 0, 1, NSIM, 0.f, 4096.f);

  // 6) masked reduction + 7) scalar loss
  zero_u32_kernel<<<1, 32, 0, stream>>>((unsigned*)(statef + 2), 4);
  accum_kernel<<<512, 256, 0, stream>>>(d2, ag, statef, statef + 2);
  final_kernel<<<1, 1, 0, stream>>>(statef + 2, out);
}